// FocalLoss_17274358464523
// MI455X (gfx1250) — compile-verified
//
#include <hip/hip_runtime.h>
#include <hip/hip_bf16.h>

// RetinaNet focal loss for MI455X (gfx1250, wave32).
// Memory-bound: ~325 MB of input -> ~14 us floor @ 23.3 TB/s.
// Single fused streaming kernel + deterministic final reduce (no atomics).
// Cross-lane reductions use V_WMMA_F32_16X16X4_F32 (exact f32 matrix path).

typedef __attribute__((ext_vector_type(2))) float v2f;
typedef __attribute__((ext_vector_type(8))) float v8f;

#define FL_ALPHA 0.25f
#define FL_THR   0.1f
#define TA   64     // anchors per block (A = 120000 = 1875 * 64)
#define BLK  256    // threads per block = 8 wave32
#define MAXM 64     // LDS capacity for annotations (M = 32 in practice)

// Exact f32 wave32 sum via WMMA:
//   A(16x4): lane L contributes x at (m = L%16, k = 2*(L/16)), 0 in the other slot.
//   B = ones. D[m][n] = x_m + x_{m+16}  (replicated over n).
//   sum of the 8 D VGPRs = half-wave sum (lanes 0-15: rows 0-7, lanes 16-31: rows 8-15);
//   one xor-16 shuffle adds the halves -> full wave sum in every lane.
// NOTE: WMMA requires EXEC all-ones: call from all 256 threads, blocks always full.
__device__ __forceinline__ float wave_sum_wmma(float x) {
    v2f a; a[0] = x;    a[1] = 0.0f;
    v2f o; o[0] = 1.0f; o[1] = 1.0f;
    v8f c = {};
    v8f d = __builtin_amdgcn_wmma_f32_16x16x4_f32(
        /*neg_a=*/false, a, /*neg_b=*/false, o,
        /*c_mod=*/(short)0, c, /*reuse_a=*/false, /*reuse_b=*/false);
    float s = d[0] + d[1] + d[2] + d[3] + d[4] + d[5] + d[6] + d[7];
    s += __shfl_xor(s, 16, 32);
    return s;
}

__global__ void __launch_bounds__(BLK)
retina_loss_main(const float* __restrict__ logits,   // [B,A,C] probabilities
                 const float* __restrict__ regr,     // [B,A,4]
                 const float* __restrict__ anchors,  // [A,4]
                 const float* __restrict__ annot,    // [B,M,5]
                 int A, int C, int M, int nBlocksPerImg,
                 float* __restrict__ clsPart,
                 float* __restrict__ regPart,
                 float* __restrict__ posPart)
{
    __shared__ float       sAnn[MAXM * 5];
    __shared__ signed char sTarg[TA];
    __shared__ float       sRed[8 * 3];

    const int blk = blockIdx.x;
    const int b   = blk / nBlocksPerImg;
    const int a0  = (blk - b * nBlocksPerImg) * TA;
    const int tid = threadIdx.x;

    // ---- stage this image's annotations (M*5 = 160 floats) ----
    if (tid < M * 5) sAnn[tid] = annot[(size_t)b * M * 5 + tid];
    __syncthreads();

    // ---- phase 1: per-anchor assignment + smooth-L1 (threads 0..TA-1) ----
    float regSum = 0.0f;
    float posCnt = 0.0f;
    if (tid < TA) {
        const int a = a0 + tid;
        const float4 an = ((const float4*)anchors)[a];   // x1,y1,x2,y2
        const float aw = an.z - an.x, ah = an.w - an.y;
        float best = -1.0f; int bm = 0;
        for (int m = 0; m < M; ++m) {
            const float bx1 = sAnn[m*5+0], by1 = sAnn[m*5+1];
            const float bx2 = sAnn[m*5+2], by2 = sAnn[m*5+3];
            const float lab = sAnn[m*5+4];
            float iw = fminf(an.z, bx2) - fmaxf(an.x, bx1); iw = fmaxf(iw, 0.0f);
            float ih = fminf(an.w, by2) - fmaxf(an.y, by1); ih = fmaxf(ih, 0.0f);
            const float inter = iw * ih;
            const float ua = fmaxf(aw * ah + (bx2-bx1)*(by2-by1) - inter, 1e-8f);
            float iou = inter / ua;
            iou = (lab != -1.0f) ? iou : -1.0f;          // mask padded annots
            if (iou > best) { best = iou; bm = m; }      // first-max = argmax
        }
        signed char tg;
        if (best >= 0.5f) {                              // positive anchor
            tg = (signed char)(int)sAnn[bm*5+4];
            posCnt = 1.0f;
            const float bx1 = sAnn[bm*5+0], by1 = sAnn[bm*5+1];
            const float bx2 = sAnn[bm*5+2], by2 = sAnn[bm*5+3];
            const float gw = fmaxf(bx2 - bx1, 1.0f), gh = fmaxf(by2 - by1, 1.0f);
            const float gcx = bx1 + 0.5f*gw, gcy = by1 + 0.5f*gh;
            const float acx = an.x + 0.5f*aw, acy = an.y + 0.5f*ah;
            const float t0 = ((gcx - acx) / aw) * 10.0f;   // / 0.1
            const float t1 = ((gcy - acy) / ah) * 10.0f;
            const float t2 = logf(gw / aw) * 5.0f;         // / 0.2
            const float t3 = logf(gh / ah) * 5.0f;
            const float4 rp = ((const float4*)regr)[(size_t)b * A + a];
            const float d0 = fabsf(t0 - rp.x), d1 = fabsf(t1 - rp.y);
            const float d2 = fabsf(t2 - rp.z), d3 = fabsf(t3 - rp.w);
            regSum  = (d0 <= FL_THR) ? 5.0f*d0*d0 : d0 - 0.05f;   // 0.5/THR = 5
            regSum += (d1 <= FL_THR) ? 5.0f*d1*d1 : d1 - 0.05f;
            regSum += (d2 <= FL_THR) ? 5.0f*d2*d2 : d2 - 0.05f;
            regSum += (d3 <= FL_THR) ? 5.0f*d3*d3 : d3 - 0.05f;
        } else {
            tg = (best >= 0.4f) ? (signed char)-1 : (signed char)-2; // bg / ignore
        }
        sTarg[tid] = tg;
    }
    __syncthreads();

    // ---- phase 2: focal classification loss over TA*C contiguous logits ----
    // Coalesced, read-once stream: use non-temporal loads (TH=NT).
    const float* lp = logits + ((size_t)b * A + a0) * C;
    const int nElem = TA * C;                           // 5120, multiple of BLK
    float clsSum = 0.0f;
    for (int e = tid; e < nElem; e += BLK) {
        const int al = e / C;
        const int c  = e - al * C;
        const int tg = sTarg[al];
        float p = __builtin_nontemporal_load(lp + e);
        p = fminf(fmaxf(p, 1e-4f), 1.0f - 1e-4f);
        float loss;
        if (tg == -2) {                                 // ignore anchor
            loss = 0.0f;
        } else if (c == tg) {                           // target == 1
            const float q = 1.0f - p;
            loss = FL_ALPHA * q * q * (-__logf(p));
        } else {                                        // target == 0
            loss = (1.0f - FL_ALPHA) * p * p * (-__logf(1.0f - p));
        }
        clsSum += loss;
    }

    // ---- phase 3: block reduction (WMMA wave sums + LDS across 8 waves) ----
    const float wc = wave_sum_wmma(clsSum);
    const float wr = wave_sum_wmma(regSum);
    const float wp = wave_sum_wmma(posCnt);
    const int wave = tid >> 5;
    if ((tid & 31) == 0) {
        sRed[wave*3+0] = wc; sRed[wave*3+1] = wr; sRed[wave*3+2] = wp;
    }
    __syncthreads();
    if (tid == 0) {
        float tc = 0.0f, tr = 0.0f, tp = 0.0f;
        for (int w = 0; w < 8; ++w) {
            tc += sRed[w*3+0]; tr += sRed[w*3+1]; tp += sRed[w*3+2];
        }
        clsPart[blk] = tc; regPart[blk] = tr; posPart[blk] = tp;
    }
}

// Deterministic fixed-order tree sum of one partial array.
__device__ float block_sum_arr(const float* __restrict__ src, int n,
                               float* sh, int tid)
{
    float s = 0.0f;
    for (int i = tid; i < n; i += BLK) s += src[i];
    sh[tid] = s;
    __syncthreads();
    for (int off = BLK >> 1; off > 0; off >>= 1) {
        if (tid < off) sh[tid] += sh[tid + off];
        __syncthreads();
    }
    const float r = sh[0];
    __syncthreads();
    return r;
}

__global__ void __launch_bounds__(BLK)
retina_loss_final(const float* __restrict__ clsPart,
                  const float* __restrict__ regPart,
                  const float* __restrict__ posPart,
                  int nBlocksPerImg, int B, float* __restrict__ out)
{
    __shared__ float sh[BLK];
    const int tid = threadIdx.x;
    float clsMean = 0.0f, regMean = 0.0f;
    for (int b = 0; b < B; ++b) {
        const float tc = block_sum_arr(clsPart + (size_t)b * nBlocksPerImg, nBlocksPerImg, sh, tid);
        const float tr = block_sum_arr(regPart + (size_t)b * nBlocksPerImg, nBlocksPerImg, sh, tid);
        const float tp = block_sum_arr(posPart + (size_t)b * nBlocksPerImg, nBlocksPerImg, sh, tid);
        if (tp > 0.5f) {                                 // num_pos > 0 gate
            clsMean += tc / fmaxf(tp, 0.01f);            // clip(num_pos, 0.01)
            regMean += tr / fmaxf(4.0f * tp, 1.0f);      // max(num_pos*4, 1)
        }
    }
    if (tid == 0) {
        out[0] = clsMean / (float)B;
        out[1] = regMean / (float)B;
    }
}

extern "C" void kernel_launch(void* const* d_in, const int* in_sizes, int n_in,
                              void* d_out, int out_size, void* d_ws, size_t ws_size,
                              hipStream_t stream) {
    const float* logits = (const float*)d_in[0];   // [B,A,C]
    const float* regr   = (const float*)d_in[1];   // [B,A,4]
    const float* anch   = (const float*)d_in[2];   // [1,A,4]
    const float* annot  = (const float*)d_in[3];   // [B,M,5]

    const int A = in_sizes[2] / 4;                 // 120000
    const int B = in_sizes[1] / (A * 4);           // 8
    const int C = in_sizes[0] / (B * A);           // 80
    const int M = in_sizes[3] / (B * 5);           // 32
    const int nbi = A / TA;                        // 1875 (A divisible by 64)

    float* clsPart = (float*)d_ws;                 // [B*nbi]
    float* regPart = clsPart + (size_t)B * nbi;    // [B*nbi]
    float* posPart = regPart + (size_t)B * nbi;    // [B*nbi]  (~180 KB total)

    retina_loss_main<<<dim3(B * nbi), dim3(BLK), 0, stream>>>(
        logits, regr, anch, annot, A, C, M, nbi, clsPart, regPart, posPart);
    retina_loss_final<<<dim3(1), dim3(BLK), 0, stream>>>(
        clsPart, regPart, posPart, nbi, B, (float*)d_out);
}